// Encoder_16535624090170
// MI455X (gfx1250) — compile-verified
//
#include <hip/hip_runtime.h>
#include <stdint.h>

// ---------------------------------------------------------------------------
// Problem constants (from reference: N=4, S=4096, E=1024, V=32000, L=6)
// ---------------------------------------------------------------------------
#define N_     4
#define S_     4096
#define E_     1024
#define L_     6
#define NSEG   64
#define SEGLEN (S_ / NSEG)            // 64
#define TPB    256                     // 8 waves of 32; each thread owns 4 e's
#define NSE    ((size_t)N_ * S_ * E_)  // 16,777,216 floats (64 MB)
#define NPE    ((size_t)N_ * NSEG * E_)// 262,144 floats per partial array
#define NE     (N_ * E_)               // 4096
#define INV_SCALE 0.03125f             // 1 / sqrt(E) = 1/32

// ---------------------------------------------------------------------------
// CDNA5 async global->LDS path (gfx1250): builtin proto takes pointers to
// 16-byte int vectors in AS1 (global) / AS3 (LDS).  Confirmed present.
// ---------------------------------------------------------------------------
#if defined(__HIP_DEVICE_COMPILE__) && defined(__has_builtin)
#if __has_builtin(__builtin_amdgcn_global_load_async_to_lds_b128) && \
    __has_builtin(__builtin_amdgcn_s_wait_asynccnt)
#define USE_ASYNC_LDS 1
#endif
#endif
#ifndef USE_ASYNC_LDS
#define USE_ASYNC_LDS 0
#endif

#if USE_ASYNC_LDS
typedef int v4i_t __attribute__((ext_vector_type(4)));
typedef __attribute__((address_space(1))) v4i_t* gptr16_t;  // ptr to AS1 v4i
typedef __attribute__((address_space(3))) v4i_t* lptr16_t;  // ptr to AS3 v4i

static __device__ __forceinline__ void async_cp16(float* lds, const float* g) {
  // 16 bytes per lane, global -> LDS, tracked by ASYNCcnt.
  float* gnc = const_cast<float*>(g);
  __builtin_amdgcn_global_load_async_to_lds_b128(
      (gptr16_t)(v4i_t*)gnc, (lptr16_t)(v4i_t*)lds, 0, 0);
}
#endif

// ---------------------------------------------------------------------------
// Helpers
// ---------------------------------------------------------------------------
static __device__ __forceinline__ float fm(float v) {
  // feature map: exact gelu + 1   (gelu(x) = 0.5 x (1 + erf(x/sqrt(2))))
  return fmaf(0.5f * v, 1.0f + erff(v * 0.70710678118654752f), 1.0f);
}

// ---------------------------------------------------------------------------
// Kernel 1: embedding gather + per-segment column sums.
//   A[n,s,e]    = emb[x[n,s], e] ;  segs[n,g,e] = sum over segment rows
// grid (NSEG, N), 256 threads, 4 e's per thread
// ---------------------------------------------------------------------------
__global__ __launch_bounds__(TPB) void embed_seg_kernel(
    const int* __restrict__ x, const float* __restrict__ emb,
    float* __restrict__ A, float* __restrict__ segs) {
  const int tid = threadIdx.x;
  const int seg = blockIdx.x;
  const int n   = blockIdx.y;
  const int e0  = 4 * tid;

  __shared__ int toks[SEGLEN];
  if (tid < SEGLEN) toks[tid] = x[n * S_ + seg * SEGLEN + tid];
  __syncthreads();

  const size_t obase = ((size_t)n * S_ + (size_t)seg * SEGLEN) * E_ + e0;
  float ax = 0.f, ay = 0.f, az = 0.f, aw = 0.f;
  for (int j = 0; j < SEGLEN; ++j) {
    const int t = toks[j];
    float4 v = *(const float4*)(emb + (size_t)t * E_ + e0);
    *(float4*)(A + obase + (size_t)j * E_) = v;
    ax += v.x; ay += v.y; az += v.z; aw += v.w;
  }
  *(float4*)(segs + ((size_t)(n * NSEG + seg)) * E_ + e0) =
      make_float4(ax, ay, az, aw);
}

// ---------------------------------------------------------------------------
// Kernel 2: segmented scan replay with fused statistics. No ctx store!
//   gpart      = partial  sum_s cur*ctx          (rep scale)
//   DO_QK also: dpart = partial sum_s q*k, qmax/qmin of q  (q=fm(cur),k=fm(ctx))
//   tailp!=0 (layer 5): last segment writes ctx[n,S-1,:] = total column sum.
// ---------------------------------------------------------------------------
template <bool DO_QK>
__global__ __launch_bounds__(TPB) void scan_kernel(
    const float* __restrict__ cur, const float* __restrict__ segs,
    float* __restrict__ gpart, float* __restrict__ dpart,
    float* __restrict__ qmxp, float* __restrict__ qmnp,
    float* __restrict__ tailp) {
  const int tid = threadIdx.x;
  const int seg = blockIdx.x;
  const int n   = blockIdx.y;
  const int e0  = 4 * tid;

  // exclusive prefix of earlier segments (tiny, L2-resident)
  float ox = 0.f, oy = 0.f, oz = 0.f, ow = 0.f;
  for (int j = 0; j < seg; ++j) {
    float4 v = *(const float4*)(segs + ((size_t)(n * NSEG + j)) * E_ + e0);
    ox += v.x; oy += v.y; oz += v.z; ow += v.w;
  }

  const float* base = cur + ((size_t)n * S_ + (size_t)seg * SEGLEN) * E_ + e0;

  float acx = 0.f, acy = 0.f, acz = 0.f, acw = 0.f;
  float gx = 0.f, gy = 0.f, gz = 0.f, gw = 0.f;
  float dx = 0.f, dy = 0.f, dz = 0.f, dw = 0.f;
  float mx = -3.0e38f, my = -3.0e38f, mz = -3.0e38f, mw = -3.0e38f;
  float nx = 3.0e38f, ny = 3.0e38f, nz = 3.0e38f, nw = 3.0e38f;

  for (int j = 0; j < SEGLEN; ++j) {
    __builtin_prefetch(base + (size_t)(j + 4) * E_, 0, 3);  // global_prefetch_b8
    float4 v = *(const float4*)(base + (size_t)j * E_);
    acx += v.x; acy += v.y; acz += v.z; acw += v.w;
    float cx = ox + acx, cy = oy + acy, cz = oz + acz, cw = ow + acw;
    gx = fmaf(v.x, cx, gx); gy = fmaf(v.y, cy, gy);
    gz = fmaf(v.z, cz, gz); gw = fmaf(v.w, cw, gw);
    if (DO_QK) {
      float qx = fm(v.x), qy = fm(v.y), qz = fm(v.z), qw = fm(v.w);
      float kx = fm(cx), ky = fm(cy), kz = fm(cz), kw = fm(cw);
      dx = fmaf(qx, kx, dx); dy = fmaf(qy, ky, dy);
      dz = fmaf(qz, kz, dz); dw = fmaf(qw, kw, dw);
      mx = fmaxf(mx, qx); my = fmaxf(my, qy);
      mz = fmaxf(mz, qz); mw = fmaxf(mw, qw);
      nx = fminf(nx, qx); ny = fminf(ny, qy);
      nz = fminf(nz, qz); nw = fminf(nw, qw);
    }
  }
  const size_t pb = ((size_t)(n * NSEG + seg)) * E_ + e0;
  *(float4*)(gpart + pb) = make_float4(gx, gy, gz, gw);
  if (DO_QK) {
    *(float4*)(dpart + pb) = make_float4(dx, dy, dz, dw);
    *(float4*)(qmxp + pb)  = make_float4(mx, my, mz, mw);
    *(float4*)(qmnp + pb)  = make_float4(nx, ny, nz, nw);
  }
  if (tailp != nullptr && seg == NSEG - 1) {   // ctx[n, S-1, :]
    *(float4*)(tailp + (size_t)n * E_ + e0) =
        make_float4(ox + acx, oy + acy, oz + acz, ow + acw);
  }
}

// ---------------------------------------------------------------------------
// Kernel 3 (layer 0 only): replay scan of embedded to rebuild ctx0 on the fly,
// write cur0 = g_e * embedded, and produce q/k statistics.  (CTX never stored.)
// ---------------------------------------------------------------------------
__global__ __launch_bounds__(TPB) void qkstat0_kernel(
    const float* __restrict__ A, const float* __restrict__ segs,
    const float* __restrict__ gvec, float* __restrict__ cur0,
    float* __restrict__ dpart, float* __restrict__ qmxp,
    float* __restrict__ qmnp) {
  const int tid = threadIdx.x;
  const int seg = blockIdx.x;
  const int n   = blockIdx.y;
  const int e0  = 4 * tid;

  float ox = 0.f, oy = 0.f, oz = 0.f, ow = 0.f;
  for (int j = 0; j < seg; ++j) {
    float4 v = *(const float4*)(segs + ((size_t)(n * NSEG + j)) * E_ + e0);
    ox += v.x; oy += v.y; oz += v.z; ow += v.w;
  }

  float4 g4 = *(const float4*)(gvec + (size_t)n * E_ + e0);
  const size_t rb = ((size_t)n * S_ + (size_t)seg * SEGLEN) * E_ + e0;

  float acx = 0.f, acy = 0.f, acz = 0.f, acw = 0.f;
  float dx = 0.f, dy = 0.f, dz = 0.f, dw = 0.f;
  float mx = -3.0e38f, my = -3.0e38f, mz = -3.0e38f, mw = -3.0e38f;
  float nx = 3.0e38f, ny = 3.0e38f, nz = 3.0e38f, nw = 3.0e38f;

  for (int j = 0; j < SEGLEN; ++j) {
    __builtin_prefetch(A + rb + (size_t)(j + 4) * E_, 0, 3);
    float4 a = *(const float4*)(A + rb + (size_t)j * E_);
    acx += a.x; acy += a.y; acz += a.z; acw += a.w;
    float cx = ox + acx, cy = oy + acy, cz = oz + acz, cw = ow + acw;
    float vx = g4.x * a.x, vy = g4.y * a.y, vz = g4.z * a.z, vw = g4.w * a.w;
    *(float4*)(cur0 + rb + (size_t)j * E_) = make_float4(vx, vy, vz, vw);
    float qx = fm(vx), qy = fm(vy), qz = fm(vz), qw = fm(vw);
    float kx = fm(cx), ky = fm(cy), kz = fm(cz), kw = fm(cw);
    dx = fmaf(qx, kx, dx); dy = fmaf(qy, ky, dy);
    dz = fmaf(qz, kz, dz); dw = fmaf(qw, kw, dw);
    mx = fmaxf(mx, qx); my = fmaxf(my, qy);
    mz = fmaxf(mz, qz); mw = fmaxf(mw, qw);
    nx = fminf(nx, qx); ny = fminf(ny, qy);
    nz = fminf(nz, qz); nw = fminf(nw, qw);
  }
  const size_t pb = ((size_t)(n * NSEG + seg)) * E_ + e0;
  *(float4*)(dpart + pb) = make_float4(dx, dy, dz, dw);
  *(float4*)(qmxp + pb)  = make_float4(mx, my, mz, mw);
  *(float4*)(qmnp + pb)  = make_float4(nx, ny, nz, nw);
}

// ---------------------------------------------------------------------------
// Kernel 4: softmax denominator  Z[n,e] = sum_s exp(c*q - m)  (partials).
// Pure streaming read pass -> staged through LDS with the CDNA5 async copy
// engine, double-buffered 8-row chunks (each lane consumes only the bytes it
// requested; s_wait_asynccnt alone orders the data, no barrier needed).
// ---------------------------------------------------------------------------
__global__ __launch_bounds__(TPB) void zsum_kernel(
    const float* __restrict__ cur, const float* __restrict__ cvec,
    const float* __restrict__ mvec, float* __restrict__ zpart) {
  const int tid = threadIdx.x;
  const int seg = blockIdx.x;
  const int n   = blockIdx.y;
  const int e0  = 4 * tid;

  float4 c4 = *(const float4*)(cvec + (size_t)n * E_ + e0);
  float4 m4 = *(const float4*)(mvec + (size_t)n * E_ + e0);
  const float* base = cur + ((size_t)n * S_ + (size_t)seg * SEGLEN) * E_;

  float zx = 0.f, zy = 0.f, zz = 0.f, zw = 0.f;
#if USE_ASYNC_LDS
  __shared__ float tile[2][8][E_];  // 2 x 32KB double buffer
  for (int r = 0; r < 8; ++r)
    async_cp16(&tile[0][r][e0], base + (size_t)r * E_ + e0);
  for (int ch = 0; ch < 8; ++ch) {
    if (ch < 7) {
      const float* nb = base + (size_t)(ch + 1) * 8 * E_;
      for (int r = 0; r < 8; ++r)
        async_cp16(&tile[(ch + 1) & 1][r][e0], nb + (size_t)r * E_ + e0);
      __builtin_amdgcn_s_wait_asynccnt(8);  // chunk ch done (in-order returns)
    } else {
      __builtin_amdgcn_s_wait_asynccnt(0);
    }
    for (int r = 0; r < 8; ++r) {
      float4 v = *(const float4*)&tile[ch & 1][r][e0];
      zx += expf(fmaf(c4.x, fm(v.x), -m4.x));
      zy += expf(fmaf(c4.y, fm(v.y), -m4.y));
      zz += expf(fmaf(c4.z, fm(v.z), -m4.z));
      zw += expf(fmaf(c4.w, fm(v.w), -m4.w));
    }
  }
#else
  for (int j = 0; j < SEGLEN; ++j) {
    float4 v = *(const float4*)(base + (size_t)j * E_ + e0);
    zx += expf(fmaf(c4.x, fm(v.x), -m4.x));
    zy += expf(fmaf(c4.y, fm(v.y), -m4.y));
    zz += expf(fmaf(c4.z, fm(v.z), -m4.z));
    zw += expf(fmaf(c4.w, fm(v.w), -m4.w));
  }
#endif
  *(float4*)(zpart + ((size_t)(n * NSEG + seg)) * E_ + e0) =
      make_float4(zx, zy, zz, zw);
}

// ---------------------------------------------------------------------------
// Tiny per-(n,e) reductions over the NSEG partials (fixed order => exact
// determinism, no float atomics).
// ---------------------------------------------------------------------------
__global__ __launch_bounds__(TPB) void reduce_g_kernel(
    const float* __restrict__ gpart, float* __restrict__ gvec) {
  const int id = blockIdx.x * TPB + threadIdx.x;   // 0..NE-1
  const int n = id >> 10, e = id & (E_ - 1);
  const size_t b = (size_t)n * NSEG * E_ + e;
  float g = 0.f;
  for (int j = 0; j < NSEG; ++j) g += gpart[b + (size_t)j * E_];
  gvec[id] = g;
}

template <bool WITH_G>
__global__ __launch_bounds__(TPB) void reduce_qk_kernel(
    const float* __restrict__ dpart, const float* __restrict__ qmxp,
    const float* __restrict__ qmnp, const float* __restrict__ gpart,
    float* __restrict__ cvec, float* __restrict__ mvec,
    float* __restrict__ gvec) {
  const int id = blockIdx.x * TPB + threadIdx.x;
  const int n = id >> 10, e = id & (E_ - 1);
  const size_t b = (size_t)n * NSEG * E_ + e;
  float d = 0.f, qmx = -3.0e38f, qmn = 3.0e38f, g = 0.f;
  for (int j = 0; j < NSEG; ++j) {
    const size_t o = b + (size_t)j * E_;
    d += dpart[o];
    qmx = fmaxf(qmx, qmxp[o]);
    qmn = fminf(qmn, qmnp[o]);
    if (WITH_G) g += gpart[o];
  }
  const float c = d * INV_SCALE;                 // energy = c * q
  cvec[id] = c;
  mvec[id] = (c >= 0.f) ? c * qmx : c * qmn;     // max_s energy (q > 0)
  if (WITH_G) gvec[id] = g;
}

__global__ __launch_bounds__(TPB) void reduce_z_kernel(
    const float* __restrict__ zpart, float* __restrict__ zinv) {
  const int id = blockIdx.x * TPB + threadIdx.x;
  const int n = id >> 10, e = id & (E_ - 1);
  const size_t b = (size_t)n * NSEG * E_ + e;
  float z = 0.f;
  for (int j = 0; j < NSEG; ++j) z += zpart[b + (size_t)j * E_];
  zinv[id] = 1.0f / z;
}

// ---------------------------------------------------------------------------
// Kernel 5: attention + residual + layernorm, fused with next layer's
// segment sums.  One block per (n, segment); loops its 64 rows; per-column
// constants live in registers across the whole segment; column sums of the
// layernorm OUTPUT are accumulated in registers and written once at the end
// (this is what the next layer's scan needs -> the seg_sum pass disappears).
//   att = exp(c*q - m) * zinv * (g * cur)      (rep = g (.) cur, g=1 layer 0)
//   y   = att + cur ;  out = LN(y) * w + b
// ---------------------------------------------------------------------------
__global__ __launch_bounds__(TPB) void row_ln_seg_kernel(
    const float* __restrict__ cur, int use_g, const float* __restrict__ gvec,
    const float* __restrict__ cvec, const float* __restrict__ mvec,
    const float* __restrict__ zivec, const float* __restrict__ lnw,
    const float* __restrict__ lnb, float* __restrict__ outp,
    float* __restrict__ segs_out) {
  const int tid = threadIdx.x;
  const int seg = blockIdx.x, n = blockIdx.y;
  const int e0 = 4 * tid;
  const size_t ce = (size_t)n * E_ + e0;

  float4 c4 = *(const float4*)(cvec + ce);
  float4 m4 = *(const float4*)(mvec + ce);
  float4 zi = *(const float4*)(zivec + ce);
  float4 w4 = *(const float4*)(lnw + e0);
  float4 b4 = *(const float4*)(lnb + e0);
  float gx = 1.f, gy = 1.f, gz = 1.f, gw = 1.f;
  if (use_g) {
    float4 g4 = *(const float4*)(gvec + ce);
    gx = g4.x; gy = g4.y; gz = g4.z; gw = g4.w;
  }

  __shared__ float red[18];
  const size_t rb0 = ((size_t)n * S_ + (size_t)seg * SEGLEN) * E_ + e0;
  float sx = 0.f, sy = 0.f, sz = 0.f, sw = 0.f;   // column sums of output

  for (int r = 0; r < SEGLEN; ++r) {
    const size_t rb = rb0 + (size_t)r * E_;
    __builtin_prefetch(cur + rb + 2 * E_, 0, 3);
    float4 v = *(const float4*)(cur + rb);

    float yx = expf(fmaf(c4.x, fm(v.x), -m4.x)) * zi.x * (gx * v.x) + v.x;
    float yy = expf(fmaf(c4.y, fm(v.y), -m4.y)) * zi.y * (gy * v.y) + v.y;
    float yz = expf(fmaf(c4.z, fm(v.z), -m4.z)) * zi.z * (gz * v.z) + v.z;
    float yw = expf(fmaf(c4.w, fm(v.w), -m4.w)) * zi.w * (gw * v.w) + v.w;

    float s1 = yx + yy + yz + yw;
    float s2 = yx * yx + yy * yy + yz * yz + yw * yw;
    for (int o = 16; o > 0; o >>= 1) {   // wave32 reduction
      s1 += __shfl_down(s1, o, 32);
      s2 += __shfl_down(s2, o, 32);
    }
    const int w = tid >> 5;
    if ((tid & 31) == 0) { red[w] = s1; red[8 + w] = s2; }
    __syncthreads();
    if (tid == 0) {
      float a = 0.f, b = 0.f;
      for (int i = 0; i < 8; ++i) { a += red[i]; b += red[8 + i]; }
      const float mu  = a * (1.0f / E_);
      const float var = b * (1.0f / E_) - mu * mu;
      red[16] = mu;
      red[17] = rsqrtf(var + 1e-5f);
    }
    __syncthreads();
    const float mu = red[16], rs = red[17];

    float4 o;
    o.x = (yx - mu) * rs * w4.x + b4.x;
    o.y = (yy - mu) * rs * w4.y + b4.y;
    o.z = (yz - mu) * rs * w4.z + b4.z;
    o.w = (yw - mu) * rs * w4.w + b4.w;
    *(float4*)(outp + rb) = o;
    sx += o.x; sy += o.y; sz += o.z; sw += o.w;
  }

  if (segs_out != nullptr) {
    *(float4*)(segs_out + ((size_t)(n * NSEG + seg)) * E_ + e0) =
        make_float4(sx, sy, sz, sw);
  }
}

// ---------------------------------------------------------------------------
// Host orchestration. Workspace (floats, ~71 MB):
//   A[NSE] | segs,gpart,dpart,qmxp,qmnp,zpart[NPE each] | gvec,cvec,mvec,zinv[NE]
// cur ping-pong: cur0=d_out -> A -> d_out -> A -> d_out -> A -> d_out(final)
// ---------------------------------------------------------------------------
extern "C" void kernel_launch(void* const* d_in, const int* in_sizes, int n_in,
                              void* d_out, int out_size, void* d_ws,
                              size_t ws_size, hipStream_t stream) {
  (void)in_sizes; (void)n_in; (void)out_size; (void)ws_size;
  const int*   x   = (const int*)d_in[0];     // tokens [N,S]
  const float* emb = (const float*)d_in[2];   // [V,E]
  const float* lnw = (const float*)d_in[3];   // [L,E]
  const float* lnb = (const float*)d_in[4];   // [L,E]
  float* outp = (float*)d_out;
  float* ws   = (float*)d_ws;

  float* A     = ws;
  float* segs  = ws + NSE;
  float* gpart = segs + NPE;
  float* dpart = gpart + NPE;
  float* qmxp  = dpart + NPE;
  float* qmnp  = qmxp + NPE;
  float* zprt  = qmnp + NPE;
  float* gvec  = zprt + NPE;
  float* cvec  = gvec + NE;
  float* mvec  = cvec + NE;
  float* ziv   = mvec + NE;

  const dim3 colg(NSEG, N_);
  const dim3 redg(NE / TPB);

  // ---- init + layer 0 ----
  embed_seg_kernel<<<colg, TPB, 0, stream>>>(x, emb, A, segs);
  scan_kernel<false><<<colg, TPB, 0, stream>>>(A, segs, gpart, dpart, qmxp,
                                               qmnp, nullptr);
  reduce_g_kernel<<<redg, TPB, 0, stream>>>(gpart, gvec);
  qkstat0_kernel<<<colg, TPB, 0, stream>>>(A, segs, gvec, outp, dpart, qmxp,
                                           qmnp);
  reduce_qk_kernel<false><<<redg, TPB, 0, stream>>>(dpart, qmxp, qmnp, gpart,
                                                    cvec, mvec, gvec);
  zsum_kernel<<<colg, TPB, 0, stream>>>(outp, cvec, mvec, zprt);
  reduce_z_kernel<<<redg, TPB, 0, stream>>>(zprt, ziv);
  // layer 0: rep = cur (use_g = 0); also emits segs for layer 1's scan
  row_ln_seg_kernel<<<colg, TPB, 0, stream>>>(outp, 0, gvec, cvec, mvec, ziv,
                                              lnw, lnb, A, segs);

  // ---- layers 1..5 ----
  for (int l = 1; l < L_; ++l) {
    const float* cin = (l & 1) ? A : outp;
    float*      cout = (l & 1) ? outp : A;
    float* tail = (l == L_ - 1) ? (outp + NSE) : nullptr;  // ctx[:, -1, :]
    scan_kernel<true><<<colg, TPB, 0, stream>>>(cin, segs, gpart, dpart, qmxp,
                                                qmnp, tail);
    reduce_qk_kernel<true><<<redg, TPB, 0, stream>>>(dpart, qmxp, qmnp, gpart,
                                                     cvec, mvec, gvec);
    zsum_kernel<<<colg, TPB, 0, stream>>>(cin, cvec, mvec, zprt);
    reduce_z_kernel<<<redg, TPB, 0, stream>>>(zprt, ziv);
    float* segs_out = (l < L_ - 1) ? segs : nullptr;  // feeds next scan
    row_ln_seg_kernel<<<colg, TPB, 0, stream>>>(cin, 1, gvec, cvec, mvec, ziv,
                                                lnw + l * E_, lnb + l * E_,
                                                cout, segs_out);
  }
}